// SSGCN_73512660238641
// MI455X (gfx1250) — compile-verified
//
#include <hip/hip_runtime.h>
#include <hip/hip_bf16.h>

#define D 128
#define K_HOPS 16
#define ALPHA 0.05f

typedef __attribute__((ext_vector_type(2))) float v2f;
typedef __attribute__((ext_vector_type(8))) float v8f;

// ---------------- degree / norm ----------------

__global__ void deg_init_kernel(float* __restrict__ deg, int n) {
    int i = blockIdx.x * blockDim.x + threadIdx.x;
    if (i < n) deg[i] = 1.0f;  // self-loop contribution
}

__global__ void deg_edges_kernel(const int* __restrict__ dst, float* __restrict__ deg, int e) {
    int i = blockIdx.x * blockDim.x + threadIdx.x;
    if (i < e) atomicAdd(&deg[dst[i]], 1.0f);
}

__global__ void dinv_kernel(const float* __restrict__ deg, float* __restrict__ dinv, int n) {
    int i = blockIdx.x * blockDim.x + threadIdx.x;
    if (i < n) {
        float d = deg[i];
        dinv[i] = (d > 0.0f) ? rsqrtf(d) : 0.0f;
    }
}

// ---------------- init: x = emb, h = alpha*emb, y = 0 ----------------

__global__ void init_kernel(const float* __restrict__ emb, float* __restrict__ x,
                            float* __restrict__ y, float* __restrict__ h, int total) {
    int i = blockIdx.x * blockDim.x + threadIdx.x;
    if (i < total) {
        float v = emb[i];
        x[i] = v;
        y[i] = 0.0f;
        h[i] = ALPHA * v;
    }
}

// ---------------- one hop: edge scatter (wave per edge, float4 per lane) ----------------

__global__ void edge_scatter_kernel(const int* __restrict__ src, const int* __restrict__ dst,
                                    const float* __restrict__ dinv,
                                    const float* __restrict__ x, float* __restrict__ y, int e) {
    int t = blockIdx.x * blockDim.x + threadIdx.x;
    int edge = t >> 5;
    if (edge >= e) return;
    int lane = t & 31;                 // 32 lanes * float4 = 128 channels
    int s = src[edge];
    int d = dst[edge];
    float coef = dinv[s] * dinv[d];
    const float4* xs = (const float4*)(x + (size_t)s * D);
    float4 v = xs[lane];
    float* yd = y + (size_t)d * D + lane * 4;
    atomicAdd(yd + 0, coef * v.x);
    atomicAdd(yd + 1, coef * v.y);
    atomicAdd(yd + 2, coef * v.z);
    atomicAdd(yd + 3, coef * v.w);
}

// ---------------- node update: add self-loop, accumulate h, clear x for next hop ----------------

__global__ void node_update_kernel(float* __restrict__ x, float* __restrict__ y,
                                   float* __restrict__ h, const float* __restrict__ dinv,
                                   int total) {
    int i = blockIdx.x * blockDim.x + threadIdx.x;
    if (i >= total) return;
    int node = i >> 7;                 // D = 128
    float di = dinv[node];
    float xv = x[i];
    float ynew = y[i] + di * di * xv;  // self-loop term of A_hat
    y[i] = ynew;                       // y becomes next x
    h[i] += ((1.0f - ALPHA) / (float)K_HOPS) * ynew;
    x[i] = 0.0f;                       // x becomes next (zeroed) y
}

// ---------------- final GEMM via V_WMMA_F32_16X16X4_F32 ----------------
// out[m, n] = sum_k h[m,k] * W[n,k] + bias[n]
// One wave owns a full 16-row stripe: 8 resident 16x16 accumulators, A fragment
// loaded once per k-step and reused across all 8 column tiles (8 independent
// WMMA chains hide the WMMA RAW latency).

__global__ void gemm_wmma_kernel(const float* __restrict__ H, const float* __restrict__ W,
                                 const float* __restrict__ bias, float* __restrict__ out,
                                 int row_tiles) {
    const int wave = threadIdx.x >> 5;
    const int lane = threadIdx.x & 31;
    const int row_tile = blockIdx.x * (blockDim.x >> 5) + wave;
    if (row_tile >= row_tiles) return;            // wave-uniform guard

    const int half = lane >> 4;                   // 0: K pair {k,k+1}, 1: {k+2,k+3}
    const int l16  = lane & 15;

    const float* arow = H + (size_t)(row_tile * 16 + l16) * D;   // A: M = l16

    v8f c[8];
    #pragma unroll
    for (int ct = 0; ct < 8; ++ct)
        c[ct] = (v8f){0.f, 0.f, 0.f, 0.f, 0.f, 0.f, 0.f, 0.f};

    #pragma unroll 2
    for (int k = 0; k < D; k += 4) {
        v2f a;
        // 16x4 f32 A layout: v0 = K = k + 2*half, v1 = K+1
        a.x = arow[k + 2 * half + 0];
        a.y = arow[k + 2 * half + 1];
        #pragma unroll
        for (int ct = 0; ct < 8; ++ct) {
            const float* bcol = W + (size_t)(ct * 16 + l16) * D; // B: N = l16 (W[n*D+k])
            v2f b;
            b.x = bcol[k + 2 * half + 0];
            b.y = bcol[k + 2 * half + 1];
            c[ct] = __builtin_amdgcn_wmma_f32_16x16x4_f32(
                        /*neg_a=*/false, a, /*neg_b=*/false, b,
                        /*c_mod=*/(short)0, c[ct], /*reuse_a=*/false, /*reuse_b=*/false);
        }
    }

    const int rbase = row_tile * 16 + half * 8;
    #pragma unroll
    for (int ct = 0; ct < 8; ++ct) {
        const int col = ct * 16 + l16;
        const float bv = bias[col];
        #pragma unroll
        for (int r = 0; r < 8; ++r)
            out[(size_t)(rbase + r) * D + col] = c[ct][r] + bv;
    }
}

// ---------------- launch ----------------

extern "C" void kernel_launch(void* const* d_in, const int* in_sizes, int n_in,
                              void* d_out, int out_size, void* d_ws, size_t ws_size,
                              hipStream_t stream) {
    const float* emb    = (const float*)d_in[0];
    const int*   eidx   = (const int*)d_in[1];
    const float* weight = (const float*)d_in[2];
    const float* bias   = (const float*)d_in[3];

    const int n = in_sizes[0] / D;     // 50000
    const int e = in_sizes[1] / 2;     // 1,600,000
    const int total = n * D;

    const int* src = eidx;
    const int* dst = eidx + e;

    // workspace layout
    float* x    = (float*)d_ws;                        // n*D
    float* y    = x + (size_t)total;                   // n*D
    float* deg  = y + (size_t)total;                   // n
    float* dinv = deg + (size_t)n;                     // n
    float* h    = (float*)d_out;                       // accumulate h in d_out

    const int B = 256;

    deg_init_kernel<<<(n + B - 1) / B, B, 0, stream>>>(deg, n);
    deg_edges_kernel<<<(e + B - 1) / B, B, 0, stream>>>(dst, deg, e);
    dinv_kernel<<<(n + B - 1) / B, B, 0, stream>>>(deg, dinv, n);
    init_kernel<<<(total + B - 1) / B, B, 0, stream>>>(emb, x, y, h, total);

    long long ethreads = (long long)e * 32;
    int eblocks = (int)((ethreads + B - 1) / B);

    float* cur = x;
    float* nxt = y;
    for (int hop = 0; hop < K_HOPS; ++hop) {
        edge_scatter_kernel<<<eblocks, B, 0, stream>>>(src, dst, dinv, cur, nxt, e);
        node_update_kernel<<<(total + B - 1) / B, B, 0, stream>>>(cur, nxt, h, dinv, total);
        float* t = cur; cur = nxt; nxt = t;            // nxt was zeroed by node_update
    }

    // h (in d_out) -> scratch, then WMMA GEMM scratch -> d_out
    hipMemcpyAsync(x, h, (size_t)total * sizeof(float), hipMemcpyDeviceToDevice, stream);

    const int row_tiles = n / 16;                      // 3125 (exact)
    const int gblocks = (row_tiles + 7) / 8;           // 8 waves (row stripes) per block
    gemm_wmma_kernel<<<gblocks, 256, 0, stream>>>(x, weight, bias, (float*)d_out, row_tiles);
}